// CustomNeuron_68582037782645
// MI455X (gfx1250) — compile-verified
//
#include <hip/hip_runtime.h>

typedef __attribute__((ext_vector_type(2))) float v2f;
typedef __attribute__((ext_vector_type(4))) float v4f;
typedef __attribute__((ext_vector_type(8))) float v8f;

#define B_DIM 32768
#define F_DIM 32
#define U_DIM 256

// ---------------------------------------------------------------------------
// Kernel 1: P_in[b] = prod_f inputs[b,f]   (inputs is [B,F] row-major)
// One thread per row; 8x float4 loads = 128B contiguous per thread.
// ---------------------------------------------------------------------------
__global__ void __launch_bounds__(256)
rowprod_kernel(const float* __restrict__ in, float* __restrict__ pin) {
    int b = blockIdx.x * blockDim.x + threadIdx.x;
    const v4f* p = (const v4f*)(in + (size_t)b * F_DIM);
    float prod = 1.0f;
#pragma unroll
    for (int i = 0; i < F_DIM / 4; ++i) {
        v4f v = p[i];
        prod *= v.x * v.y * v.z * v.w;
    }
    pin[b] = prod;
}

// ---------------------------------------------------------------------------
// Kernel 2: P_w[u] = prod_f weight[f,u]    (weight is [F,U] row-major)
// One block of 256 threads; coalesced column-strided reads.
// ---------------------------------------------------------------------------
__global__ void __launch_bounds__(256)
colprod_kernel(const float* __restrict__ w, float* __restrict__ pw) {
    int u = threadIdx.x;
    float prod = 1.0f;
#pragma unroll
    for (int f = 0; f < F_DIM; ++f) prod *= w[f * U_DIM + u];
    pw[u] = prod;
}

// ---------------------------------------------------------------------------
// Kernel 3: out[b,u] = P_in[b] * P_w[u] + bias[u] via V_WMMA_F32_16X16X4_F32.
// Rank-1 update: only the K=0 slice of A (16x4) and B (4x16) is non-zero.
//   A: lanes 0-15, VGPR0 = (M=lane, K=0)  -> P_in[tileRow*16 + lane]
//   B: lanes 0-15, VGPR0 = (K=0, N=lane)  -> P_w[tileCol*16 + lane]
//   C: all 8 accumulator VGPRs = bias[u]  (bias depends only on N = lane%16)
// One wave per 16x16 output tile; 2048 x 16 tiles = 32768 waves exactly.
// ---------------------------------------------------------------------------
__global__ void __launch_bounds__(256)
outer_wmma_kernel(const float* __restrict__ pin,
                  const float* __restrict__ pw,
                  const float* __restrict__ bias,
                  float* __restrict__ out) {
    int wave    = (int)((blockIdx.x * blockDim.x + threadIdx.x) >> 5);
    int lane    = threadIdx.x & 31;
    int tileRow = wave >> 4;   // 0..2047  (b tiles of 16)
    int tileCol = wave & 15;   // 0..15    (u tiles of 16)
    int n       = lane & 15;
    int u       = tileCol * 16 + n;

    float pwv = pw[u];
    float bv  = bias[u];

    // A (16x4 f32): K=0 column carries P_in, everything else zero.
    v2f a;
    a.x = (lane < 16) ? pin[tileRow * 16 + lane] : 0.0f;
    a.y = 0.0f;

    // B (4x16 f32): K=0 row carries P_w, everything else zero.
    v2f bm;
    bm.x = (lane < 16) ? pwv : 0.0f;
    bm.y = 0.0f;

    // C: broadcast bias along M.
    v8f c;
#pragma unroll
    for (int i = 0; i < 8; ++i) c[i] = bv;

    // 8 args: (neg_a, A, neg_b, B, c_mod, C, reuse_a, reuse_b)
    v8f d = __builtin_amdgcn_wmma_f32_16x16x4_f32(
        false, a, false, bm, (short)0, c, false, false);

    // D layout: VGPR i -> row M=i (lanes 0-15) / M=i+8 (lanes 16-31), N=lane%16.
    int mbase = (lane < 16) ? 0 : 8;
#pragma unroll
    for (int i = 0; i < 8; ++i) {
        int bg = tileRow * 16 + mbase + i;
        out[(size_t)bg * U_DIM + u] = d[i];
    }
}

// ---------------------------------------------------------------------------
extern "C" void kernel_launch(void* const* d_in, const int* in_sizes, int n_in,
                              void* d_out, int out_size, void* d_ws, size_t ws_size,
                              hipStream_t stream) {
    const float* inputs = (const float*)d_in[0];  // [B, F]
    const float* weight = (const float*)d_in[1];  // [F, U]
    // d_in[2] = weight_selector: dead code in the reference, unused.
    const float* bias   = (const float*)d_in[3];  // [U]
    float* out = (float*)d_out;                   // [B, U]

    float* pin = (float*)d_ws;          // B_DIM floats
    float* pw  = pin + B_DIM;           // U_DIM floats

    rowprod_kernel<<<B_DIM / 256, 256, 0, stream>>>(inputs, pin);
    colprod_kernel<<<1, 256, 0, stream>>>(weight, pw);

    // 32768 waves total -> 4096 blocks x 8 waves
    int nWaves  = (B_DIM / 16) * (U_DIM / 16);
    int nBlocks = nWaves / 8;
    outer_wmma_kernel<<<nBlocks, 256, 0, stream>>>(pin, pw, bias, out);
}